// Dihedral_19559281066776
// MI455X (gfx1250) — compile-verified
//
#include <hip/hip_runtime.h>
#include <math.h>

#define NTERMS 3
#define TYPE_STRIDE 390625   // 25^4
#define NBATCH 256
#define TILE 256

struct F3 { float x, y, z; };

__device__ __forceinline__ F3 f3sub(F3 a, F3 b) { return {a.x - b.x, a.y - b.y, a.z - b.z}; }
__device__ __forceinline__ F3 f3cross(F3 a, F3 b) {
    return {a.y * b.z - a.z * b.y,
            a.z * b.x - a.x * b.z,
            a.x * b.y - a.y * b.x};
}
__device__ __forceinline__ float f3dot(F3 a, F3 b) { return a.x * b.x + a.y * b.y + a.z * b.z; }

__device__ __forceinline__ F3 ldpos(const float* __restrict__ p, int i) {
    const float* q = p + 3 * (long long)i;
    F3 r; r.x = q[0]; r.y = q[1]; r.z = q[2];
    return r;
}

// ---- CDNA5 async global->LDS copy (gfx1250), GVS mode: saddr + vgpr offset ----
__device__ __forceinline__ void async_ld_b32(unsigned lds_addr, unsigned goff, const void* base) {
    asm volatile("global_load_async_to_lds_b32 %0, %1, %2"
                 :: "v"(lds_addr), "v"(goff), "s"(base) : "memory");
}
__device__ __forceinline__ void wait_async_le5() {
    asm volatile("s_wait_asynccnt 0x5" ::: "memory");
}
__device__ __forceinline__ void wait_async_0() {
    asm volatile("s_wait_asynccnt 0x0" ::: "memory");
}

// Stage one tile (4 mapping rows + batch ids) into LDS buffer `buf`.
// Offsets are clamped per-lane so tail tiles stay uniform (EXEC all-ones).
__device__ __forceinline__ void issue_tile(int base, int n_dih, unsigned sbase, int buf,
                                           const int* __restrict__ mapping,
                                           const int* __restrict__ mapping_batch) {
    int i  = base + (int)threadIdx.x;
    int ii = (i < n_dih) ? i : (n_dih - 1);
    unsigned goff = (unsigned)ii * 4u;
    unsigned l = sbase + (unsigned)buf * (5u * TILE * 4u) + (unsigned)threadIdx.x * 4u;
    async_ld_b32(l,               goff, (const void*)mapping);
    async_ld_b32(l + 1u*TILE*4u,  goff, (const void*)(mapping + 1 * (size_t)n_dih));
    async_ld_b32(l + 2u*TILE*4u,  goff, (const void*)(mapping + 2 * (size_t)n_dih));
    async_ld_b32(l + 3u*TILE*4u,  goff, (const void*)(mapping + 3 * (size_t)n_dih));
    async_ld_b32(l + 4u*TILE*4u,  goff, (const void*)mapping_batch);
}

extern "C" __global__ void dihedral_zero_kernel(float* __restrict__ out, int n) {
    int i = blockIdx.x * blockDim.x + threadIdx.x;
    if (i < n) out[i] = 0.0f;
}

// Fold thetas/ks into per-type-tuple records: {k_n*cos(th0_n)}x3, sum(k) | {k_n*sin(th0_n)}x3, 0
extern "C" __global__ void dihedral_table_kernel(const float* __restrict__ thetas,
                                                 const float* __restrict__ ks,
                                                 float* __restrict__ tab, int n) {
    int i = blockIdx.x * blockDim.x + threadIdx.x;
    if (i >= n) return;
    float A[NTERMS], B[NTERMS], ksum = 0.0f;
#pragma unroll
    for (int t = 0; t < NTERMS; ++t) {
        float th = thetas[t * TYPE_STRIDE + i];
        float k  = ks[t * TYPE_STRIDE + i];
        float s, c;
        __sincosf(th, &s, &c);
        A[t] = k * c; B[t] = k * s; ksum += k;
    }
    float4* o = (float4*)(tab + (size_t)i * 8);
    o[0] = make_float4(A[0], A[1], A[2], ksum);
    o[1] = make_float4(B[0], B[1], B[2], 0.0f);
}

template <bool USE_TAB>
__global__ __launch_bounds__(256)
void dihedral_main(const float* __restrict__ pos,
                   const float* __restrict__ thetas,
                   const float* __restrict__ ks,
                   const int*   __restrict__ mapping,       // (4, n_dih)
                   const int*   __restrict__ atom_types,
                   const int*   __restrict__ mapping_batch, // sorted
                   const float4* __restrict__ tab,          // TYPE_STRIDE x 2 float4
                   float*       __restrict__ energy,        // NBATCH
                   int n_dih)
{
    __shared__ float bins[NBATCH];
    __shared__ int   stage[2][5 * TILE];   // double-buffered index staging (10 KB)

    for (int t = threadIdx.x; t < NBATCH; t += blockDim.x) bins[t] = 0.0f;
    __syncthreads();

    const int      stride = gridDim.x * blockDim.x;
    const int      tid    = (int)threadIdx.x;
    const unsigned lane   = threadIdx.x & 31u;
    const unsigned sbase  = (unsigned)(uintptr_t)(&stage[0][0]);

    int base0 = blockIdx.x * blockDim.x;
    int buf = 0;
    if (base0 < n_dih) issue_tile(base0, n_dih, sbase, 0, mapping, mapping_batch);

    for (int base = base0; base < n_dih; base += stride) {
        int nxt = base + stride;
        if (nxt < n_dih) {
            issue_tile(nxt, n_dih, sbase, buf ^ 1, mapping, mapping_batch);
            wait_async_le5();                 // current tile's 5 loads landed
        } else {
            wait_async_0();
        }

        const int* sp = &stage[buf][0];
        int i0 = sp[tid];
        int i1 = sp[1 * TILE + tid];
        int i2 = sp[2 * TILE + tid];
        int i3 = sp[3 * TILE + tid];
        int b  = sp[4 * TILE + tid];
        bool active = (base + tid) < n_dih;

        F3 p0 = ldpos(pos, i0);
        F3 p1 = ldpos(pos, i1);
        F3 p2 = ldpos(pos, i2);
        F3 p3 = ldpos(pos, i3);

        F3 b1 = f3sub(p1, p0);
        F3 b2 = f3sub(p2, p1);
        F3 b3 = f3sub(p3, p2);
        F3 n1 = f3cross(b1, b2);
        F3 n2 = f3cross(b2, b3);

        float rb2 = __frsqrt_rn(f3dot(b2, b2));
        F3 b2n = {b2.x * rb2, b2.y * rb2, b2.z * rb2};
        F3 m1 = f3cross(n1, b2n);

        float x = f3dot(n1, n2);
        float y = f3dot(m1, n2);

        int t0 = atom_types[i0];
        int t1 = atom_types[i1];
        int t2 = atom_types[i2];
        int t3 = atom_types[i3];
        int idx = ((t0 * 25 + t1) * 25 + t2) * 25 + t3;

        float V;
        if (USE_TAB) {
            // cos/sin(theta) directly from geometry; nθ by angle addition (no atan2/cos)
            float h2 = fmaf(x, x, y * y);
            float r  = (h2 > 0.0f) ? __frsqrt_rn(h2) : 0.0f;
            float c1 = (h2 > 0.0f) ? x * r : 1.0f;
            float s1 = y * r;
            float c2 = fmaf(2.0f * c1, c1, -1.0f);
            float s2 = 2.0f * s1 * c1;
            float c3 = c1 * c2 - s1 * s2;
            float s3 = s1 * c2 + c1 * s2;
            float4 a  = tab[(size_t)idx * 2];       // {k cos th0} x3, ksum
            float4 bq = tab[(size_t)idx * 2 + 1];   // {k sin th0} x3
            V = a.w - (a.x * c1 + a.y * c2 + a.z * c3
                     + bq.x * s1 + bq.y * s2 + bq.z * s3);
        } else {
            float theta = atan2f(y, x);
            V = 0.0f;
#pragma unroll
            for (int n = 0; n < NTERMS; ++n) {
                float th0 = thetas[n * TYPE_STRIDE + idx];
                float kk  = ks[n * TYPE_STRIDE + idx];
                float arg = fmaf((float)(n + 1), theta, -th0);
                V = fmaf(kk, 1.0f - __cosf(arg), V);
            }
        }
        if (!active) V = 0.0f;

        // wave32 segmented inclusive scan (batch ids sorted -> long runs)
#pragma unroll
        for (int off = 1; off < 32; off <<= 1) {
            float vo = __shfl_up(V, off, 32);
            int   bo = __shfl_up(b, off, 32);
            if ((int)lane >= off && bo == b) V += vo;
        }
        int bnext = __shfl_down(b, 1, 32);
        if (lane == 31u || bnext != b) atomicAdd(&bins[b], V);  // ds_add_f32

        buf ^= 1;
    }

    __syncthreads();
    for (int t = threadIdx.x; t < NBATCH; t += blockDim.x) {
        float v = bins[t];
        if (v != 0.0f) atomicAdd(&energy[t], v);                // global_atomic_add_f32
    }
}

extern "C" void kernel_launch(void* const* d_in, const int* in_sizes, int n_in,
                              void* d_out, int out_size, void* d_ws, size_t ws_size,
                              hipStream_t stream)
{
    const float* pos           = (const float*)d_in[0];
    const float* thetas        = (const float*)d_in[1];
    const float* ks            = (const float*)d_in[2];
    const int*   mapping       = (const int*)d_in[3];
    const int*   atom_types    = (const int*)d_in[4];
    const int*   mapping_batch = (const int*)d_in[5];
    float*       energy        = (float*)d_out;

    int n_dih = in_sizes[3] / 4;

    dihedral_zero_kernel<<<(out_size + 255) / 256, 256, 0, stream>>>(energy, out_size);

    int blocks = 2048;
    int need   = (n_dih + TILE - 1) / TILE;
    if (need < blocks) blocks = need;
    if (blocks < 1) blocks = 1;

    size_t tab_bytes = (size_t)TYPE_STRIDE * 8 * sizeof(float);   // 12.5 MB
    if (ws_size >= tab_bytes) {
        float* tab = (float*)d_ws;
        dihedral_table_kernel<<<(TYPE_STRIDE + 255) / 256, 256, 0, stream>>>(
            thetas, ks, tab, TYPE_STRIDE);
        dihedral_main<true><<<blocks, 256, 0, stream>>>(
            pos, thetas, ks, mapping, atom_types, mapping_batch,
            (const float4*)tab, energy, n_dih);
    } else {
        dihedral_main<false><<<blocks, 256, 0, stream>>>(
            pos, thetas, ks, mapping, atom_types, mapping_batch,
            nullptr, energy, n_dih);
    }
}